// ComponentSeparationLoss_81346680586855
// MI455X (gfx1250) — compile-verified
//
#include <hip/hip_runtime.h>
#include <hip/hip_bf16.h>

#define BB   4
#define HH   256
#define WW   256
#define CCH  128
#define NN   32
#define HWPX (HH*WW)          // 65536 pixels per batch
#define BIGL (HWPX)           // background sentinel, matches reference
#define KSEL 52428u           // 0-indexed order statistic: 0.8*(HWPX-1), exact
#define CCL_ITERS 20

typedef __attribute__((ext_vector_type(2))) float v2f;
typedef __attribute__((ext_vector_type(8))) float v8f;

// ---------------------------------------------------------------------------
// Phase 1: gather center feature rows + their squared norms. 128 threads total.
// ---------------------------------------------------------------------------
__global__ void k_gather(const float* __restrict__ feat, const int* __restrict__ centers,
                         float* __restrict__ gt, float* __restrict__ gnorm) {
    int t = threadIdx.x;              // 0..127
    int b = t >> 5, n = t & 31;
    int r = centers[(b * NN + n) * 2 + 0];
    int c = centers[(b * NN + n) * 2 + 1];
    const float* src = feat + ((size_t)b * HWPX + (size_t)r * WW + c) * CCH;
    float* dst = gt + (size_t)(b * NN + n) * CCH;
    float s = 0.f;
    for (int k = 0; k < CCH; k += 4) {
        float4 v = *(const float4*)(src + k);
        *(float4*)(dst + k) = v;
        s += v.x * v.x + v.y * v.y + v.z * v.z + v.w * v.w;
    }
    gnorm[b * NN + n] = s;
}

// ---------------------------------------------------------------------------
// Phase 2: WMMA distance GEMM + min over centers.
// One wave = 16-pixel x 32-center tile: two f32 16x16 accumulators,
// K-loop over 128 in steps of 4 using V_WMMA_F32_16X16X4_F32.
// A layout (16x4 f32): lane L holds row M=L&15; VGPR0/1 = K {0,1}+2*(L>>4).
// B layout (4x16 f32): lane L holds col N=L&15; VGPR0/1 = K {0,1}+2*(L>>4).
// C/D layout: VGPR v, lane L -> element (M = v + 8*(L>>4), N = L&15).
// ---------------------------------------------------------------------------
__global__ void __launch_bounds__(256) k_distmin(const float* __restrict__ feat,
                                                 const float* __restrict__ gt,
                                                 const float* __restrict__ gnorm,
                                                 float* __restrict__ contrast) {
    int wave  = (blockIdx.x * blockDim.x + threadIdx.x) >> 5;
    int lane  = threadIdx.x & 31;
    int b     = wave >> 12;            // 4096 tiles per batch
    int tile  = wave & 4095;
    int pbase = tile << 4;             // 16 pixels per tile
    int lo = lane & 15, hi = lane >> 4;

    const float* rowA  = feat + ((size_t)b * HWPX + pbase + lo) * CCH;
    const float* rowB0 = gt + (size_t)(b * NN + lo) * CCH;        // centers 0..15
    const float* rowB1 = gt + (size_t)(b * NN + 16 + lo) * CCH;   // centers 16..31

    v8f acc0 = {0.f,0.f,0.f,0.f,0.f,0.f,0.f,0.f};
    v8f acc1 = {0.f,0.f,0.f,0.f,0.f,0.f,0.f,0.f};
    float pn = 0.f;                    // partial |pixel|^2 (this lane's K-range)

    for (int k = 0; k < CCH; k += 4) {
        int off = k + 2 * hi;
        v2f a  = *(const v2f*)(rowA  + off);
        v2f b0 = *(const v2f*)(rowB0 + off);
        v2f b1 = *(const v2f*)(rowB1 + off);
        pn += a[0] * a[0] + a[1] * a[1];
        acc0 = __builtin_amdgcn_wmma_f32_16x16x4_f32(false, a, false, b0,
                                                     (short)0, acc0, false, false);
        acc1 = __builtin_amdgcn_wmma_f32_16x16x4_f32(false, a, false, b1,
                                                     (short)0, acc1, false, false);
    }
    // full |pixel|^2: lane L and L^16 cover complementary K halves of pixel L&15
    pn += __shfl_xor(pn, 16, 32);
    float gn0 = gnorm[b * NN + lo];
    float gn1 = gnorm[b * NN + 16 + lo];

#pragma unroll
    for (int v = 0; v < 8; ++v) {
        int pl = v + 8 * hi;                         // local pixel of this C/D row
        float pnv = __shfl(pn, pl, 32);              // lane pl holds pixel pl's norm
        float d0 = pnv + gn0 - 2.f * acc0[v];
        float d1 = pnv + gn1 - 2.f * acc1[v];
        float m = fminf(sqrtf(fmaxf(d0, 0.f)), sqrtf(fmaxf(d1, 0.f)));
        // min over the 16 lanes of this half-wave (all 32 centers covered)
        m = fminf(m, __shfl_xor(m, 1, 32));
        m = fminf(m, __shfl_xor(m, 2, 32));
        m = fminf(m, __shfl_xor(m, 4, 32));
        m = fminf(m, __shfl_xor(m, 8, 32));
        if (lo == 0) contrast[(size_t)b * HWPX + pbase + pl] = m;
    }
}

// ---------------------------------------------------------------------------
// Phase 3: exact 80th percentile = order statistic KSEL via 4-pass radix select
// on the (monotone, non-negative) float bit pattern.
// ---------------------------------------------------------------------------
__global__ void k_selinit(unsigned* prefix, unsigned* krem, unsigned* hist) {
    int t = blockIdx.x * blockDim.x + threadIdx.x;
    for (int i = t; i < BB * 4 * 256; i += blockDim.x * gridDim.x) hist[i] = 0u;
    if (t < BB) { prefix[t] = 0u; krem[t] = KSEL; }
}

__global__ void k_hist(const float* __restrict__ contrast, const unsigned* __restrict__ prefix,
                       unsigned* __restrict__ hist, int pass) {
    __shared__ unsigned h[256];
    int tid = threadIdx.x;
    h[tid] = 0u;
    __syncthreads();
    int b = blockIdx.x >> 8;                         // 256 blocks per batch
    int idx = ((blockIdx.x & 255) << 8) + tid;
    unsigned bits = __float_as_uint(contrast[(size_t)b * HWPX + idx]);
    int shift = 24 - 8 * pass;
    bool match = (pass == 0) || ((bits >> (shift + 8)) == prefix[b]);
    if (match) atomicAdd(&h[(bits >> shift) & 255u], 1u);
    __syncthreads();
    if (h[tid]) atomicAdd(&hist[(b * 4 + pass) * 256 + tid], h[tid]);
}

__global__ void k_select(unsigned* prefix, unsigned* krem, const unsigned* __restrict__ hist,
                         int pass) {
    if (threadIdx.x != 0) return;
    int b = blockIdx.x;
    const unsigned* h = hist + (b * 4 + pass) * 256;
    unsigned pref = prefix[b], kr = krem[b], c = 0u;
    for (int bin = 0; bin < 256; ++bin) {
        unsigned cnt = h[bin];
        if (kr < c + cnt) { prefix[b] = (pref << 8) | (unsigned)bin; krem[b] = kr - c; return; }
        c += cnt;
    }
}

// ---------------------------------------------------------------------------
// Phase 4: mask + union-find CCL (8-connected), fixed iteration count.
// ---------------------------------------------------------------------------
__global__ void k_maskinit(const float* __restrict__ contrast, const unsigned* __restrict__ prefix,
                           int* __restrict__ labels) {
    int i = blockIdx.x * blockDim.x + threadIdx.x;   // 0 .. B*HWPX-1
    int b = i >> 16;
    float thr = __uint_as_float(prefix[b]);
    int p = i & (HWPX - 1);
    labels[i] = (contrast[i] >= thr) ? p : BIGL;
}

__device__ __forceinline__ int find_root(const int* lab, int l) {
    int v = lab[l];
    while (v < l) { l = v; v = lab[l]; }             // labels[x] <= x always
    return l;
}

__global__ void k_merge(int* __restrict__ labels) {
    int i = blockIdx.x * blockDim.x + threadIdx.x;
    int b = i >> 16, p = i & (HWPX - 1);
    int* lab = labels + (b << 16);
    if (lab[p] == BIGL) return;
    int y = p >> 8, x = p & 255;
    int rp = find_root(lab, p);
    int nmin = rp;
    for (int dy = -1; dy <= 1; ++dy)
        for (int dx = -1; dx <= 1; ++dx) {
            if (dy == 0 && dx == 0) continue;
            int yy = y + dy, xx = x + dx;
            if ((unsigned)yy < HH && (unsigned)xx < WW) {
                int q = (yy << 8) | xx;
                if (lab[q] != BIGL) {
                    int rq = find_root(lab, q);
                    nmin = nmin < rq ? nmin : rq;
                }
            }
        }
    if (nmin < rp) atomicMin(&lab[rp], nmin);
}

__global__ void k_compress(int* __restrict__ labels) {
    int i = blockIdx.x * blockDim.x + threadIdx.x;
    int b = i >> 16, p = i & (HWPX - 1);
    int* lab = labels + (b << 16);
    if (lab[p] == BIGL) return;
    lab[p] = find_root(lab, p);
}

// ---------------------------------------------------------------------------
// Phase 5: count same-component center pairs -> merge rate.
// ---------------------------------------------------------------------------
__global__ void k_count(const int* __restrict__ labels, const int* __restrict__ centers,
                        float* __restrict__ out) {
    __shared__ int ids[BB * NN];
    __shared__ unsigned cnt;
    int t = threadIdx.x;                             // 0..127
    int b = t >> 5, n = t & 31;
    int r = centers[(b * NN + n) * 2 + 0];
    int c = centers[(b * NN + n) * 2 + 1];
    ids[t] = labels[(b << 16) + (r << 8) + c];
    if (t == 0) cnt = 0u;
    __syncthreads();
    unsigned local = 0u;
    int id = ids[t];
    if (id != BIGL)
        for (int j = n + 1; j < NN; ++j)
            if (ids[(b << 5) | j] == id) ++local;
    atomicAdd(&cnt, local);
    __syncthreads();
    if (t == 0) out[0] = (float)cnt / 1984.0f;       // B * N*(N-1)/2 = 1984
}

// ---------------------------------------------------------------------------
extern "C" void kernel_launch(void* const* d_in, const int* in_sizes, int n_in,
                              void* d_out, int out_size, void* d_ws, size_t ws_size,
                              hipStream_t stream) {
    (void)in_sizes; (void)n_in; (void)out_size; (void)ws_size;
    const float* feat    = (const float*)d_in[0];    // [B,H,W,C] f32
    const int*   centers = (const int*)d_in[1];      // [B,N,2] int32
    float*       out     = (float*)d_out;            // scalar f32

    // workspace layout (~2.2 MB)
    float*    gt       = (float*)d_ws;                       // B*N*C
    float*    gnorm    = gt + (size_t)BB * NN * CCH;         // B*N
    unsigned* prefix   = (unsigned*)(gnorm + BB * NN);       // B
    unsigned* krem     = prefix + BB;                        // B
    unsigned* hist     = krem + BB;                          // B*4*256
    float*    contrast = (float*)(hist + BB * 4 * 256);      // B*HWPX
    int*      labels   = (int*)(contrast + (size_t)BB * HWPX); // B*HWPX

    const int nthr = 256;
    const int nblk = (BB * HWPX) / nthr;                     // 1024

    k_gather<<<1, 128, 0, stream>>>(feat, centers, gt, gnorm);
    k_distmin<<<(BB * HWPX / 16) * 32 / nthr, nthr, 0, stream>>>(feat, gt, gnorm, contrast);
    k_selinit<<<4, 256, 0, stream>>>(prefix, krem, hist);
    for (int p = 0; p < 4; ++p) {
        k_hist<<<nblk, nthr, 0, stream>>>(contrast, prefix, hist, p);
        k_select<<<BB, 32, 0, stream>>>(prefix, krem, hist, p);
    }
    k_maskinit<<<nblk, nthr, 0, stream>>>(contrast, prefix, labels);
    for (int it = 0; it < CCL_ITERS; ++it) {
        k_merge<<<nblk, nthr, 0, stream>>>(labels);
        k_compress<<<nblk, nthr, 0, stream>>>(labels);
    }
    k_count<<<1, 128, 0, stream>>>(labels, centers, out);
}